// VanillaRNN_56143812493420
// MI455X (gfx1250) — compile-verified
//
#include <hip/hip_runtime.h>
#include <hip/hip_bf16.h>

// MI455X / gfx1250, wave32. RNN scan: per-step GEMM (128x2048x2048 f32) on the
// matrix pipe via V_WMMA_F32_16X16X4_F32 (full-precision f32 matrix op).
// Wt pre-packed into B-fragment order once (16 MB in d_ws); h row-block staged
// in LDS in 32KB K-chunks. 256 sequential step launches on `stream` provide
// the scan dependency (graph-capture safe).

#define T_DIM 256
#define B_DIM 128
#define H_DIM 2048
#define KC    512                 // K-chunk staged in LDS (16 x 512 f32 = 32 KB)
#define JT_COUNT (H_DIM / 16)     // 128 column tiles

typedef float v2f __attribute__((ext_vector_type(2)));
typedef float v8f __attribute__((ext_vector_type(8)));

// ---------------------------------------------------------------------------
// Pack Wt (H x H row-major) into WMMA B-fragment order.
// For K-tile kt (k = 4*kt .. 4*kt+3) and column tile jt, lane L holds
// B[k][n] for n = jt*16 + (L&15), k = 4*kt + 2*(L>>4) + v  (v = 0,1 per VGPR),
// where B[k][n] = Wt[n][k].  Layout: Bpack[((kt*JT_COUNT + jt)*32 + L)*2 + v].
// ---------------------------------------------------------------------------
__global__ __launch_bounds__(256) void prep_pack_kernel(
    const float* __restrict__ Wt, float* __restrict__ Bpack) {
  size_t tid  = (size_t)blockIdx.x * 256 + threadIdx.x;   // 0 .. H*H/2-1
  int    L    = (int)(tid & 31);
  size_t rest = tid >> 5;
  int    jt   = (int)(rest % JT_COUNT);
  size_t kt   = rest / JT_COUNT;
  int    j    = jt * 16 + (L & 15);
  int    k    = (int)(kt * 4) + ((L >> 4) << 1);
  size_t src  = (size_t)j * H_DIM + k;
  v2f w;
  w.x = Wt[src];
  w.y = Wt[src + 1];
  *(v2f*)&Bpack[tid * 2] = w;
}

// ---------------------------------------------------------------------------
// h0 = tanh(hidden_init); cbias = bt + be; pred0 = h0 . Wp + bp  (scalar).
// ---------------------------------------------------------------------------
__global__ __launch_bounds__(256) void prep_h0_kernel(
    const float* __restrict__ hinit, const float* __restrict__ bt,
    const float* __restrict__ be, const float* __restrict__ Wp,
    const float* __restrict__ bp, float* __restrict__ h0,
    float* __restrict__ cbias, float* __restrict__ pred0) {
  __shared__ float red[256];
  int tid = threadIdx.x;
  float part = 0.0f;
  for (int k = tid; k < H_DIM; k += 256) {
    float h = tanhf(hinit[k]);
    h0[k] = h;
    cbias[k] = bt[k] + be[k];
    part += h * Wp[k];
  }
  red[tid] = part;
  __syncthreads();
  for (int off = 128; off > 0; off >>= 1) {
    if (tid < off) red[tid] += red[tid + off];
    __syncthreads();
  }
  if (tid == 0) pred0[0] = red[0] + bp[0];
}

__global__ void init_preds_kernel(float* __restrict__ preds,
                                  const float* __restrict__ bp) {
  int i = blockIdx.x * blockDim.x + threadIdx.x;
  if (i < T_DIM * B_DIM) preds[i] = bp[0];
}

// ---------------------------------------------------------------------------
// One RNN step. Grid (16, 8): blockIdx.x = column block (128 cols),
// blockIdx.y = row block (16 batch rows). 256 threads = 8 wave32s; wave w
// owns the 16x16 tile at columns (cb*8 + w)*16.
//
// pre[b][j] = sum_k h[b][k]*Wt[j][k] + (bt[j]+be[j]) + We[j]*s_b
//   with s_b = g_b*obs_b + (1-g_b)*pred_b
// h_new = tanh(pre); pred_new[b] = sum_j h_new[b][j]*Wp[j] + bp (bp pre-init).
// ---------------------------------------------------------------------------
__global__ __launch_bounds__(256) void rnn_step_kernel(
    const float* __restrict__ hprev, int hstride,       // hstride 0 => broadcast h0
    const float* __restrict__ predprev, int pstride,    // pstride 0 => scalar pred0
    const float* __restrict__ obs_row,                  // obs  + t*B
    const float* __restrict__ gate_row,                 // gate + t*B
    const float* __restrict__ Bpack,
    const float* __restrict__ cbias,
    const float* __restrict__ We,
    const float* __restrict__ Wp,
    float* __restrict__ hs_t,                           // hs   + t*B*H
    float* __restrict__ pres_t,                         // pres + t*B*H
    float* __restrict__ preds_t)                        // preds + t*B
{
  __shared__ __align__(16) float sh_h[16 * KC];
  __shared__ float sh_s[16];
  __shared__ float sh_pred[16];

  const int tid     = threadIdx.x;
  const int lane    = tid & 31;
  const int wave    = tid >> 5;              // 0..7
  const int cb      = blockIdx.x;            // 0..15
  const int rb      = blockIdx.y;            // 0..7
  const int rowBase = rb * 16;
  const int jt      = cb * 8 + wave;         // 0..127
  const int colBase = jt * 16;

  if (tid < 16) {
    int b = rowBase + tid;
    float g = gate_row[b];
    float o = obs_row[b];
    float p = predprev[b * pstride];
    sh_s[tid]    = g * o + (1.0f - g) * p;
    sh_pred[tid] = 0.0f;
  }

  // Fragment addressing (16x16x4 f32 WMMA):
  //  A: lane L -> M = L&15, K = 2*(L>>4) + v      (v in VGPR 0/1)
  //  B: lane L -> N = L&15, K = 2*(L>>4) + v
  const int n    = lane & 15;
  const int half = lane >> 4;
  const int aoff = (lane & 15) * KC + (half << 1);

  const float*  bptr    = Bpack + (((size_t)jt) * 32 + lane) * 2;
  const size_t  bstride = (size_t)JT_COUNT * 64;   // floats per kt step (= 8192)

  v8f acc = {};

  for (int kc = 0; kc < H_DIM; kc += KC) {
    __syncthreads();   // previous chunk fully consumed (also covers sh_s init)
    // Cooperative stage of h rows [rowBase, rowBase+16) x cols [kc, kc+KC)
    for (int i = tid; i < 16 * (KC / 4); i += 256) {
      int r  = i / (KC / 4);
      int c4 = i - r * (KC / 4);
      const float4* src =
          (const float4*)(hprev + (size_t)(rowBase + r) * hstride + kc) + c4;
      ((float4*)&sh_h[r * KC])[c4] = *src;
    }
    __syncthreads();

#pragma unroll 4
    for (int k0 = 0; k0 < KC; k0 += 4) {
      v2f a = *(const v2f*)&sh_h[aoff + k0];
      v2f b = *(const v2f*)bptr;
      bptr += bstride;
      acc = __builtin_amdgcn_wmma_f32_16x16x4_f32(
          /*neg_a=*/false, a, /*neg_b=*/false, b,
          /*c_mod=*/(short)0, acc, /*reuse_a=*/false, /*reuse_b=*/false);
    }
  }

  // Epilogue: fuse biases / gate-embedding, tanh, stores, pred reduction.
  // C/D layout: VGPR r, lane L -> M = r + 8*(L>>4), N = L&15.
  const int   j   = colBase + n;
  const float we  = We[j];
  const float cbj = cbias[j];
  const float wpj = Wp[j];

#pragma unroll
  for (int r = 0; r < 8; ++r) {
    const int   rloc = r + (half << 3);           // row within 16-row block
    const int   b    = rowBase + rloc;
    const float pre  = acc[r] + cbj + we * sh_s[rloc];
    const float hn   = tanhf(pre);
    const size_t oidx = (size_t)b * H_DIM + j;
    pres_t[oidx] = pre;
    hs_t[oidx]   = hn;
    // pred partial: reduce hn*Wp[j] over the 16 N-lanes of each half-wave.
    float pp = hn * wpj;
    pp += __shfl_xor(pp, 8, 32);
    pp += __shfl_xor(pp, 4, 32);
    pp += __shfl_xor(pp, 2, 32);
    pp += __shfl_xor(pp, 1, 32);
    if (n == 0) atomicAdd(&sh_pred[rloc], pp);    // ds_add_f32
  }
  __syncthreads();
  if (tid < 16) atomicAdd(&preds_t[rowBase + tid], sh_pred[tid]);
}

// ---------------------------------------------------------------------------
extern "C" void kernel_launch(void* const* d_in, const int* in_sizes, int n_in,
                              void* d_out, int out_size, void* d_ws,
                              size_t ws_size, hipStream_t stream) {
  (void)in_sizes; (void)n_in; (void)out_size; (void)ws_size;

  const float* obs   = (const float*)d_in[0];   // (T,B)
  const float* gate  = (const float*)d_in[1];   // (T,B)
  const float* hinit = (const float*)d_in[2];   // (1,H)
  const float* Wt    = (const float*)d_in[3];   // (H,H)
  const float* bt    = (const float*)d_in[4];   // (H,)
  const float* We    = (const float*)d_in[5];   // (H,1) -> H
  const float* be    = (const float*)d_in[6];   // (H,)
  const float* Wp    = (const float*)d_in[7];   // (1,H) -> H
  const float* bp    = (const float*)d_in[8];   // (1,)

  float* hs    = (float*)d_out;                                // T*B*H
  float* preds = hs + (size_t)T_DIM * B_DIM * H_DIM;           // T*B
  float* pres  = preds + (size_t)T_DIM * B_DIM;                // T*B*H

  // Workspace layout (needs ~16.02 MB of d_ws):
  float* Bpack = (float*)d_ws;                                 // H*H
  float* h0    = Bpack + (size_t)H_DIM * H_DIM;                // H
  float* cbias = h0 + H_DIM;                                   // H
  float* pred0 = cbias + H_DIM;                                // 1

  // One-time prep (re-run every call for determinism; inputs are read-only).
  prep_pack_kernel<<<(unsigned)((size_t)H_DIM * H_DIM / 2 / 256), 256, 0,
                     stream>>>(Wt, Bpack);
  prep_h0_kernel<<<1, 256, 0, stream>>>(hinit, bt, be, Wp, bp, h0, cbias, pred0);
  init_preds_kernel<<<(T_DIM * B_DIM + 255) / 256, 256, 0, stream>>>(preds, bp);

  dim3 grid(16, 8);
  for (int t = 0; t < T_DIM; ++t) {
    const float* hprev = (t == 0) ? h0 : (hs + (size_t)(t - 1) * B_DIM * H_DIM);
    const int    hstr  = (t == 0) ? 0 : H_DIM;
    const float* pprev = (t == 0) ? pred0 : (preds + (size_t)(t - 1) * B_DIM);
    const int    pstr  = (t == 0) ? 0 : 1;
    rnn_step_kernel<<<grid, 256, 0, stream>>>(
        hprev, hstr, pprev, pstr,
        obs + (size_t)t * B_DIM, gate + (size_t)t * B_DIM,
        Bpack, cbias, We, Wp,
        hs + (size_t)t * B_DIM * H_DIM,
        pres + (size_t)t * B_DIM * H_DIM,
        preds + (size_t)t * B_DIM);
  }
}